// Encoder_19559281066222
// MI455X (gfx1250) — compile-verified
//
#include <hip/hip_runtime.h>

typedef __attribute__((ext_vector_type(2))) float v2f;
typedef __attribute__((ext_vector_type(8))) float v8f;

#define N_NODES 100000
#define N_EDGES 500000
#define K_DIM   128
#define O_DIM   512
#define M_BLK   32      // rows per workgroup (2 M-tiles of 16)
#define LDS_STR 132     // padded A-tile stride (conflict-free column reads)

// ---------------- kernel 1: zero deg + agg workspace ----------------
__global__ void zero_ws_kernel(float4* __restrict__ ws, long n4) {
  long i = (long)blockIdx.x * blockDim.x + threadIdx.x;
  long stride = (long)gridDim.x * blockDim.x;
  for (; i < n4; i += stride) ws[i] = make_float4(0.f, 0.f, 0.f, 0.f);
}

// ---------------- kernel 2: degree on destination nodes ----------------
__global__ void degree_kernel(const int* __restrict__ ei, float* __restrict__ deg) {
  int e = blockIdx.x * blockDim.x + threadIdx.x;
  if (e < N_EDGES) {
    int dst = ei[N_EDGES + e];
    atomicAdd(&deg[dst], 1.0f);
  }
}

// ---------------- kernel 3: dinv = rsqrt(deg + 1)  (+1 = self loop) ----------------
__global__ void dinv_kernel(const float* __restrict__ deg, float* __restrict__ dinv) {
  int i = blockIdx.x * blockDim.x + threadIdx.x;
  if (i < N_NODES) {
    float d = deg[i] + 1.0f;
    dinv[i] = __frsqrt_rn(d);
  }
}

// ---------------- kernel 4: normalized scatter of raw features (one wave per edge) ----------------
__global__ void scatter_kernel(const int* __restrict__ ei, const float* __restrict__ x,
                               const float* __restrict__ dinv, float* __restrict__ agg) {
  int wave = (int)((blockIdx.x * blockDim.x + threadIdx.x) >> 5);
  int lane = threadIdx.x & 31;
  if (wave >= N_EDGES) return;
  int src = ei[wave];
  int dst = ei[N_EDGES + wave];
  float coef = dinv[src] * dinv[dst];
  const float* xs = x + (long)src * K_DIM;
  float* ad = agg + (long)dst * K_DIM;
#pragma unroll
  for (int t = 0; t < 4; ++t) {
    int c = lane + t * 32;
    atomicAdd(&ad[c], xs[c] * coef);
  }
}

// ---------------- kernel 5: (agg + dinv^2 * x) @ W + b, PReLU, via f32 WMMA ----------------
// Block: 256 threads = 8 waves. Block tile: 32 rows x 512 cols.
// Wave tile: 2 M-tiles x 4 N-tiles (32 rows x 64 cols) -> each B fragment feeds 2 WMMAs.
__global__ __launch_bounds__(256) void gcn_gemm_kernel(
    const float* __restrict__ agg, const float* __restrict__ x,
    const float* __restrict__ dinv, const float* __restrict__ W,
    const float* __restrict__ bias, const float* __restrict__ pw,
    float* __restrict__ out) {
  __shared__ float sA[M_BLK * LDS_STR];  // 32 x 128 fused A tile

  const int tid = threadIdx.x;
  const int rowbase = blockIdx.x * M_BLK;  // N_NODES % 32 == 0, no tail

  // Cooperative fused A-tile load: A = agg + dinv^2 * x   (self-loop folded in)
#pragma unroll
  for (int t = 0; t < (M_BLK * K_DIM) / 256; ++t) {
    int idx = t * 256 + tid;       // 0..4095
    int r = idx >> 7;              // 0..31
    int c = idx & 127;             // 0..127
    long g = rowbase + r;
    float di = dinv[g];
    sA[r * LDS_STR + c] = agg[g * K_DIM + c] + di * di * x[g * K_DIM + c];
  }
  __syncthreads();

  const int wave = tid >> 5;
  const int lane = tid & 31;
  const int half = lane >> 4;   // 0: K offsets {0,1}; 1: K offsets {2,3}
  const int l15  = lane & 15;   // A: row M; B/C/D: col N
  const int cb0  = wave * 64;   // this wave's first output column

  v8f acc[2][4] = {};           // [m-tile][n-tile] 16x16 f32 accumulators (64 VGPRs)

  const float* sArow0 = &sA[l15 * LDS_STR];            // M-tile 0: rows 0..15
  const float* sArow1 = &sA[(16 + l15) * LDS_STR];     // M-tile 1: rows 16..31
#pragma unroll
  for (int kk = 0; kk < K_DIM / 4; ++kk) {
    const int kb = kk * 4 + half * 2;  // K base for this lane's half
    // A fragments: 16x4 f32 -> v2f per lane (ISA 7.12.2), 8B-aligned LDS reads
    v2f a0 = *(const v2f*)(&sArow0[kb]);
    v2f a1 = *(const v2f*)(&sArow1[kb]);
    const float* w0 = &W[(long)kb * O_DIM + cb0 + l15];
#pragma unroll
    for (int t = 0; t < 4; ++t) {
      // B fragment: 4x16 f32; VGPR0 = row kb, VGPR1 = row kb+1 (lane half picks K pair)
      v2f b = { w0[t * 16], w0[O_DIM + t * 16] };
      acc[0][t] = __builtin_amdgcn_wmma_f32_16x16x4_f32(
          false, a0, false, b, (short)0, acc[0][t], false, false);
      acc[1][t] = __builtin_amdgcn_wmma_f32_16x16x4_f32(
          false, a1, false, b, (short)0, acc[1][t], false, false);
    }
  }

  // Epilogue: bias + channel-wise PReLU, write straight to d_out
#pragma unroll
  for (int mt = 0; mt < 2; ++mt) {
#pragma unroll
    for (int t = 0; t < 4; ++t) {
      int col = cb0 + t * 16 + l15;
      float bc = bias[col];
      float pc = pw[col];
#pragma unroll
      for (int r = 0; r < 8; ++r) {
        int m = rowbase + mt * 16 + r + half * 8;  // C/D layout: VGPR r -> row r (+8 upper half)
        float o = acc[mt][t][r] + bc;
        out[(long)m * O_DIM + col] = (o >= 0.0f) ? o : pc * o;
      }
    }
  }
}

extern "C" void kernel_launch(void* const* d_in, const int* in_sizes, int n_in,
                              void* d_out, int out_size, void* d_ws, size_t ws_size,
                              hipStream_t stream) {
  (void)in_sizes; (void)n_in; (void)out_size; (void)ws_size;
  const float* x  = (const float*)d_in[0];
  const int*   ei = (const int*)d_in[1];   // [2, E] row-major: src = ei[0:E], dst = ei[E:2E]
  const float* W  = (const float*)d_in[2]; // [128, 512] row-major
  const float* b  = (const float*)d_in[3];
  const float* pw = (const float*)d_in[4];
  float* out = (float*)d_out;

  float* ws   = (float*)d_ws;
  float* deg  = ws;                     // [N]
  float* dinv = ws + N_NODES;           // [N]
  float* agg  = ws + 2 * N_NODES;       // [N, 128]  (offsets are 16B-aligned)

  const long zero_f = 2L * N_NODES + (long)N_NODES * K_DIM;  // 13,000,000 (divisible by 4)
  zero_ws_kernel<<<2048, 256, 0, stream>>>((float4*)ws, zero_f / 4);
  degree_kernel<<<(N_EDGES + 255) / 256, 256, 0, stream>>>(ei, deg);
  dinv_kernel<<<(N_NODES + 255) / 256, 256, 0, stream>>>(deg, dinv);
  scatter_kernel<<<N_EDGES / 8, 256, 0, stream>>>(ei, x, dinv, agg);   // 8 edges (waves) per block
  gcn_gemm_kernel<<<N_NODES / M_BLK, 256, 0, stream>>>(agg, x, dinv, W, b, pw, out);
}